// SelfAttentionModule_85658827751728
// MI455X (gfx1250) — compile-verified
//
#include <hip/hip_runtime.h>
#include <hip/hip_bf16.h>

typedef __attribute__((ext_vector_type(2))) float v2f;
typedef __attribute__((ext_vector_type(8))) float v8f;

#define WMMA_F32(a, b, c) \
  __builtin_amdgcn_wmma_f32_16x16x4_f32(false, (a), false, (b), (short)0, (c), false, false)

constexpr int Bn = 4;     // batch
constexpr int Cc = 512;   // channels
constexpr int Aa = 64;    // attention features
constexpr int Nn = 4096;  // H*W tokens

// ---------------------------------------------------------------------------
// Kernel 1: fused 1x1-conv projections  Q = xT*Wg+bg, K = xT*Wf+bf, V = xT*Wh+bh
// Grid: (N/64, B), 256 threads (8 waves). Each WG computes a 64-token slab.
// ---------------------------------------------------------------------------
__global__ __launch_bounds__(256)
void proj_kernel(const float* __restrict__ x,
                 const float* __restrict__ Wf, const float* __restrict__ bf,
                 const float* __restrict__ Wg, const float* __restrict__ bg,
                 const float* __restrict__ Wh, const float* __restrict__ bh,
                 float* __restrict__ Q, float* __restrict__ K, float* __restrict__ V)
{
  __shared__ float xs[128 * 64];  // x chunk: 128 channels x 64 tokens (32 KB)

  const int b    = blockIdx.y;
  const int n0   = blockIdx.x * 64;
  const int tid  = threadIdx.x;
  const int w    = tid >> 5;        // wave id (wave32)
  const int lane = tid & 31;
  const int cl   = lane & 15;       // matrix column / A-row within fragment
  const int hi   = lane >> 4;       // half-wave select
  const int loff = hi << 1;         // K sub-offset (0 or 2)
  const int rowhi = hi << 3;        // +8 rows in C/D fragments

  // Tile ownership: Q/K 4x4 tiles -> wave w owns tiles 2w,2w+1.
  //                 V 4x32 tiles  -> wave w owns columns [64w, 64w+64), all 4 row tiles.
  v8f qacc[2], kacc[2], vacc[16];
#pragma unroll
  for (int i = 0; i < 2; ++i) {
    const int ct = (w * 2 + i) & 3;
    const float bq = bg[ct * 16 + cl];
    const float bk = bf[ct * 16 + cl];
#pragma unroll
    for (int r = 0; r < 8; ++r) { qacc[i][r] = bq; kacc[i][r] = bk; }
  }
#pragma unroll
  for (int rt = 0; rt < 4; ++rt)
#pragma unroll
    for (int c4 = 0; c4 < 4; ++c4) {
      const float bv = bh[w * 64 + c4 * 16 + cl];
#pragma unroll
      for (int r = 0; r < 8; ++r) vacc[rt * 4 + c4][r] = bv;
    }

  for (int cc = 0; cc < Cc; cc += 128) {
    __syncthreads();
    {  // stage x[b, cc..cc+127, n0..n0+63] (rows contiguous, float4 coalesced)
      const float4* src = (const float4*)(x + ((size_t)b * Cc + cc) * Nn + n0);
      float4* dst = (float4*)xs;
      for (int idx = tid; idx < 128 * 16; idx += 256) {
        const int r = idx >> 4, c4i = idx & 15;
        dst[idx] = src[(size_t)r * (Nn / 4) + c4i];
      }
    }
    __syncthreads();

    for (int ks = 0; ks < 128; ks += 4) {
      const int k0 = ks + loff;     // channel within chunk for this half-wave
      const int kg = cc + k0;       // global channel
      // ---- Q and K tiles (2 each per wave) ----
#pragma unroll
      for (int i = 0; i < 2; ++i) {
        const int t = w * 2 + i, rt = t >> 2, ct = t & 3;
        v2f a;  a.x = xs[k0 * 64 + rt * 16 + cl];
                a.y = xs[(k0 + 1) * 64 + rt * 16 + cl];
        v2f bg2; bg2.x = Wg[(size_t)kg * Aa + ct * 16 + cl];
                 bg2.y = Wg[(size_t)(kg + 1) * Aa + ct * 16 + cl];
        qacc[i] = WMMA_F32(a, bg2, qacc[i]);
        v2f bf2; bf2.x = Wf[(size_t)kg * Aa + ct * 16 + cl];
                 bf2.y = Wf[(size_t)(kg + 1) * Aa + ct * 16 + cl];
        kacc[i] = WMMA_F32(a, bf2, kacc[i]);
      }
      // ---- V tiles (16 per wave) ----
#pragma unroll
      for (int rt = 0; rt < 4; ++rt) {
        v2f a;  a.x = xs[k0 * 64 + rt * 16 + cl];
                a.y = xs[(k0 + 1) * 64 + rt * 16 + cl];
#pragma unroll
        for (int c4 = 0; c4 < 4; ++c4) {
          const int col = w * 64 + c4 * 16 + cl;
          v2f bb; bb.x = Wh[(size_t)kg * Cc + col];
                  bb.y = Wh[(size_t)(kg + 1) * Cc + col];
          vacc[rt * 4 + c4] = WMMA_F32(a, bb, vacc[rt * 4 + c4]);
        }
      }
    }
  }

  // ---- store Q/K/V slabs ----
#pragma unroll
  for (int i = 0; i < 2; ++i) {
    const int t = w * 2 + i, rt = t >> 2, ct = t & 3;
#pragma unroll
    for (int r = 0; r < 8; ++r) {
      const int row = n0 + rt * 16 + r + rowhi;
      Q[((size_t)b * Nn + row) * Aa + ct * 16 + cl] = qacc[i][r];
      K[((size_t)b * Nn + row) * Aa + ct * 16 + cl] = kacc[i][r];
    }
  }
#pragma unroll
  for (int rt = 0; rt < 4; ++rt)
#pragma unroll
    for (int c4 = 0; c4 < 4; ++c4)
#pragma unroll
      for (int r = 0; r < 8; ++r) {
        const int row = n0 + rt * 16 + r + rowhi;
        V[((size_t)b * Nn + row) * Cc + w * 64 + c4 * 16 + cl] = vacc[rt * 4 + c4][r];
      }
}

// ---------------------------------------------------------------------------
// Kernel 2: flash attention. Grid: (N/64, B), 256 threads (8 waves).
// Each WG owns 64 query rows; o accumulator 64x512 fp32 lives in VGPRs.
// ---------------------------------------------------------------------------
__global__ __launch_bounds__(256)
void attn_kernel(const float* __restrict__ x,
                 const float* __restrict__ Q, const float* __restrict__ K,
                 const float* __restrict__ V, const float* __restrict__ scale_p,
                 float* __restrict__ out)
{
  __shared__ float Qs[64 * 64];
  __shared__ float Ks[64 * 64];
  __shared__ float Ss[64 * 64];
  __shared__ float mrow[64], lrow[64], arow[64];
  __shared__ float pmax[4 * 64], psum[4 * 64];

  const int b    = blockIdx.y;
  const int q0   = blockIdx.x * 64;
  const int tid  = threadIdx.x;
  const int w    = tid >> 5;
  const int lane = tid & 31;
  const int cl   = lane & 15;
  const int hi   = lane >> 4;
  const int loff = hi << 1;
  const int rowhi = hi << 3;
  const int srow_id = tid & 63;   // softmax: row handled by this thread
  const int quarter = tid >> 6;   // softmax: 16-column slice within the row

  {  // load Q slab
    const float4* src = (const float4*)(Q + ((size_t)b * Nn + q0) * Aa);
    float4* dst = (float4*)Qs;
    for (int idx = tid; idx < 64 * 16; idx += 256) dst[idx] = src[idx];
  }
  if (tid < 64) { mrow[tid] = -3.0e38f; lrow[tid] = 0.0f; }

  v8f oacc[16];
#pragma unroll
  for (int t = 0; t < 16; ++t)
#pragma unroll
    for (int r = 0; r < 8; ++r) oacc[t][r] = 0.0f;

  for (int j0 = 0; j0 < Nn; j0 += 64) {
    __syncthreads();  // previous iteration fully consumed Ks/Ss
    {  // load K slab
      const float4* src = (const float4*)(K + ((size_t)b * Nn + j0) * Aa);
      float4* dst = (float4*)Ks;
      for (int idx = tid; idx < 64 * 16; idx += 256) dst[idx] = src[idx];
    }
    // prefetch next j-block's K (16 KB) and V (128 KB) into cache while we
    // compute this block (global_prefetch_b8; no counter cost)
    if (j0 + 64 < Nn) {
      const float* nk = K + ((size_t)b * Nn + j0 + 64) * Aa;
      __builtin_prefetch(nk + tid * 16, 0, 3);           // 256 x 64B = whole slab
      const float* nv = V + ((size_t)b * Nn + j0 + 64) * Cc;
#pragma unroll
      for (int pp = 0; pp < 4; ++pp)                     // 1024 x 128B lines
        __builtin_prefetch(nv + (size_t)(pp * 256 + tid) * 32, 0, 3);
    }
    __syncthreads();

    // ---- S = Q * K^T (64x64), 2 tiles per wave ----
#pragma unroll
    for (int i = 0; i < 2; ++i) {
      const int t = w * 2 + i, rt = t >> 2, ct = t & 3;
      v8f s = {};
      for (int ks = 0; ks < 64; ks += 4) {
        const int k0 = ks + loff;
        v2f a;  a.x = Qs[(rt * 16 + cl) * 64 + k0];
                a.y = Qs[(rt * 16 + cl) * 64 + k0 + 1];
        v2f bb; bb.x = Ks[(ct * 16 + cl) * 64 + k0];    // B[k][j] = K[j][k]
                bb.y = Ks[(ct * 16 + cl) * 64 + k0 + 1];
        s = WMMA_F32(a, bb, s);
      }
#pragma unroll
      for (int r = 0; r < 8; ++r)
        Ss[(rt * 16 + r + rowhi) * 64 + ct * 16 + cl] = s[r];
    }
    __syncthreads();

    // ---- online softmax, parallel over all 256 threads (4 per row) ----
    {  // phase 1: partial row max over this thread's 16-column slice
      const float* sr = Ss + srow_id * 64 + quarter * 16;
      float m = -3.0e38f;
#pragma unroll
      for (int c = 0; c < 16; ++c) m = fmaxf(m, sr[c]);
      pmax[quarter * 64 + srow_id] = m;
    }
    __syncthreads();
    if (tid < 64) {  // phase 2: combine maxes, compute rescale factor
      const float mold = mrow[tid];
      float mnew = fmaxf(fmaxf(pmax[tid], pmax[64 + tid]),
                         fmaxf(pmax[128 + tid], pmax[192 + tid]));
      mnew = fmaxf(mnew, mold);
      arow[tid] = __expf(mold - mnew);
      mrow[tid] = mnew;
    }
    __syncthreads();
    {  // phase 3: exponentiate slice in place, partial sums
      float* sr = Ss + srow_id * 64 + quarter * 16;
      const float mnew = mrow[srow_id];
      float sum = 0.0f;
#pragma unroll
      for (int c = 0; c < 16; ++c) {
        const float p = __expf(sr[c] - mnew);
        sr[c] = p;
        sum += p;
      }
      psum[quarter * 64 + srow_id] = sum;
    }
    __syncthreads();
    if (tid < 64) {  // phase 4: fold into running denominator
      lrow[tid] = lrow[tid] * arow[tid] +
                  (psum[tid] + psum[64 + tid]) + (psum[128 + tid] + psum[192 + tid]);
    }
    __syncthreads();

    // ---- rescale o, then o += P * V (V streamed from L2-resident workspace) ----
#pragma unroll
    for (int rt = 0; rt < 4; ++rt) {
      float al[8];
#pragma unroll
      for (int r = 0; r < 8; ++r) al[r] = arow[rt * 16 + r + rowhi];
#pragma unroll
      for (int c4 = 0; c4 < 4; ++c4) {
        v8f acc = oacc[rt * 4 + c4];
#pragma unroll
        for (int r = 0; r < 8; ++r) acc[r] *= al[r];
        oacc[rt * 4 + c4] = acc;
      }
      for (int ks = 0; ks < 64; ks += 4) {
        const int k0 = ks + loff;
        v2f a;  a.x = Ss[(rt * 16 + cl) * 64 + k0];
                a.y = Ss[(rt * 16 + cl) * 64 + k0 + 1];
        const float* vp = V + ((size_t)b * Nn + j0 + k0) * Cc + w * 64 + cl;
#pragma unroll
        for (int c4 = 0; c4 < 4; ++c4) {
          v2f bb; bb.x = vp[c4 * 16];
                  bb.y = vp[c4 * 16 + Cc];
          oacc[rt * 4 + c4] = WMMA_F32(a, bb, oacc[rt * 4 + c4]);
        }
      }
    }
  }
  __syncthreads();

  // ---- epilogue: y.flat = scale * o.flat + x.flat (raw reshape => same flat idx)
  const float scl = scale_p[0];
#pragma unroll
  for (int rt = 0; rt < 4; ++rt)
#pragma unroll
    for (int c4 = 0; c4 < 4; ++c4)
#pragma unroll
      for (int r = 0; r < 8; ++r) {
        const int row = rt * 16 + r + rowhi;
        const int col = w * 64 + c4 * 16 + cl;
        const float ov = oacc[rt * 4 + c4][r] / lrow[row];
        const size_t idx = (size_t)b * Nn * Cc + (size_t)(q0 + row) * Cc + col;
        out[idx] = scl * ov + x[idx];
      }
}

// ---------------------------------------------------------------------------
extern "C" void kernel_launch(void* const* d_in, const int* in_sizes, int n_in,
                              void* d_out, int out_size, void* d_ws, size_t ws_size,
                              hipStream_t stream) {
  const float* x     = (const float*)d_in[0];
  const float* Wf    = (const float*)d_in[1];
  const float* bf    = (const float*)d_in[2];
  const float* Wg    = (const float*)d_in[3];
  const float* bg    = (const float*)d_in[4];
  const float* Wh    = (const float*)d_in[5];
  const float* bh    = (const float*)d_in[6];
  const float* scale = (const float*)d_in[7];
  float* out = (float*)d_out;

  float* Qw = (float*)d_ws;                       // [4,4096,64]  4 MB
  float* Kw = Qw + (size_t)Bn * Nn * Aa;          // [4,4096,64]  4 MB
  float* Vw = Kw + (size_t)Bn * Nn * Aa;          // [4,4096,512] 32 MB

  dim3 grid(Nn / 64, Bn), block(256);
  proj_kernel<<<grid, block, 0, stream>>>(x, Wf, bf, Wg, bg, Wh, bh, Qw, Kw, Vw);
  attn_kernel<<<grid, block, 0, stream>>>(x, Qw, Kw, Vw, scale, out);
}